// CL_Loss_28707561406570
// MI455X (gfx1250) — compile-verified
//
#include <hip/hip_runtime.h>
#include <math.h>

typedef float v2f __attribute__((ext_vector_type(2)));
typedef float v8f __attribute__((ext_vector_type(8)));

#define D_DIM 512   // representation dim
#define P_DIM 256   // projection dim
#define S_DIM 64    // sequence length
#define N_ROWS 2048 // 2*B
#define INV_T 20.0f // 1/0.05

// ---------------------------------------------------------------------------
// Kernel 1: fused gather + projection.  proj[i,p] = rep[i, len_i-1, :] . W[p,:] + b[p]
// One wave (32 lanes) per 16x16 output tile; K stepped by 4 with V_WMMA_F32_16X16X4_F32.
// ---------------------------------------------------------------------------
__global__ __launch_bounds__(32)
void proj_kernel(const float* __restrict__ rep, const int* __restrict__ lens,
                 const float* __restrict__ W, const float* __restrict__ bias,
                 float* __restrict__ Z)
{
    const int lane = threadIdx.x;   // 0..31
    const int half = lane >> 4;     // 0/1  -> K sub-pair
    const int l16  = lane & 15;     // row within tile for A/B operand loads
    const int i0 = blockIdx.x * 16; // sample-row block
    const int p0 = blockIdx.y * 16; // projection-col block

    // gather: per-lane base pointer into representations
    const int i = i0 + l16;
    int sidx = lens[i] - 1;
    sidx = sidx < 0 ? 0 : (sidx > S_DIM - 1 ? S_DIM - 1 : sidx);
    const float* arow = rep + ((size_t)i * S_DIM + sidx) * D_DIM + 2 * half;
    const float* brow = W + (size_t)(p0 + l16) * D_DIM + 2 * half;

    v8f acc = {};
    for (int k0 = 0; k0 < D_DIM; k0 += 4) {
        v2f a = *(const v2f*)(arow + k0);   // 8B-aligned: rows are 2KB-aligned, +0 or +8
        v2f b = *(const v2f*)(brow + k0);
        acc = __builtin_amdgcn_wmma_f32_16x16x4_f32(
            /*neg_a=*/false, a, /*neg_b=*/false, b,
            /*c_mod=*/(short)0, acc, /*reuse_a=*/false, /*reuse_b=*/false);
    }

    const float bb = bias[p0 + l16];
    #pragma unroll
    for (int v = 0; v < 8; ++v) {
        const int row = i0 + v + 8 * half;          // C layout: VGPR v holds rows v / v+8
        Z[(size_t)row * P_DIM + p0 + l16] = acc[v] + bb;
    }
}

// ---------------------------------------------------------------------------
// Kernel 2: in-place row L2 normalization of Z (2048 x 256). One wave per row.
// ---------------------------------------------------------------------------
__global__ __launch_bounds__(256)
void norm_kernel(float* __restrict__ Z)
{
    const int lane = threadIdx.x & 31;
    const int wave = threadIdx.x >> 5;
    const int row  = blockIdx.x * 8 + wave;
    float* zr = Z + (size_t)row * P_DIM;

    float vals[8];
    float ss = 0.f;
    #pragma unroll
    for (int e = 0; e < 8; ++e) {
        vals[e] = zr[lane + 32 * e];
        ss += vals[e] * vals[e];
    }
    #pragma unroll
    for (int m = 16; m >= 1; m >>= 1) ss += __shfl_xor(ss, m, 32);

    const float s = 1.0f / fmaxf(sqrtf(ss), 1e-8f);
    #pragma unroll
    for (int e = 0; e < 8; ++e) zr[lane + 32 * e] = vals[e] * s;
}

// ---------------------------------------------------------------------------
// Kernel 3: tiled S = Z.Z^T * (1/T) with fused online logsumexp per row.
// grid.x = 128 row-blocks (16 rows), grid.y = 8 column groups (256 cols = 16 tiles).
// The A panel (16 rows x 256 K = 64 v2f per lane = 128 VGPRs) is hoisted into
// registers once and reused across all 16 column tiles, halving hot-loop vmem.
// Masked entries (c==r) use a -1e30 sentinel so the branchless online update's
// exp() underflows them away. Target logit S[r, r^1024] written once globally.
// ---------------------------------------------------------------------------
__global__ __launch_bounds__(32)
void sim_lse_kernel(const float* __restrict__ Z, float* __restrict__ partial,
                    float* __restrict__ target)
{
    const int lane = threadIdx.x;
    const int half = lane >> 4;
    const int l16  = lane & 15;
    const int i0 = blockIdx.x * 16;
    const int cg = blockIdx.y;

    const float* arow = Z + (size_t)(i0 + l16) * P_DIM + 2 * half;

    // Hoist the entire A panel for this row-block into registers (128 VGPRs).
    v2f apan[64];
    #pragma unroll
    for (int k = 0; k < 64; ++k) apan[k] = *(const v2f*)(arow + 4 * k);

    float mx[8], sm[8];
    #pragma unroll
    for (int v = 0; v < 8; ++v) { mx[v] = -1e30f; sm[v] = 0.f; }

    for (int t = 0; t < 16; ++t) {
        const int j0 = cg * 256 + t * 16;
        const float* brow = Z + (size_t)(j0 + l16) * P_DIM + 2 * half;

        v8f acc = {};
        #pragma unroll
        for (int k = 0; k < 64; ++k) {
            v2f b = *(const v2f*)(brow + 4 * k);
            acc = __builtin_amdgcn_wmma_f32_16x16x4_f32(
                false, apan[k], false, b, (short)0, acc, false, false);
        }

        const int c = j0 + l16;
        #pragma unroll
        for (int v = 0; v < 8; ++v) {
            const int r = i0 + v + 8 * half;
            float val = (c == r) ? -1e30f : acc[v] * INV_T;
            if (c == (r ^ 1024)) target[r] = val;   // exactly one writer per row
            const float nm = fmaxf(mx[v], val);
            sm[v] = sm[v] * __expf(mx[v] - nm) + __expf(val - nm);
            mx[v] = nm;
        }
    }

    // combine the 16 lanes that hold the same row
    __shared__ float smx[16][16];
    __shared__ float ssm[16][16];
    #pragma unroll
    for (int v = 0; v < 8; ++v) {
        smx[v + 8 * half][l16] = mx[v];
        ssm[v + 8 * half][l16] = sm[v];
    }
    __syncthreads();

    if (lane < 16) {
        float M = -1e30f;
        for (int c2 = 0; c2 < 16; ++c2) M = fmaxf(M, smx[lane][c2]);
        float S = 0.f;
        for (int c2 = 0; c2 < 16; ++c2) S += ssm[lane][c2] * __expf(smx[lane][c2] - M);
        const int r = i0 + lane;
        partial[(r * 8 + cg) * 2 + 0] = M;
        partial[(r * 8 + cg) * 2 + 1] = S;
    }
}

// ---------------------------------------------------------------------------
// Kernel 4: merge the 8 per-group partials per row, loss = mean(lse - target).
// ---------------------------------------------------------------------------
__global__ __launch_bounds__(256)
void finalize_kernel(const float* __restrict__ partial, const float* __restrict__ target,
                     float* __restrict__ out)
{
    __shared__ float red[256];
    float acc = 0.f;
    for (int j = 0; j < 8; ++j) {
        const int r = threadIdx.x + 256 * j;
        float M = -1e30f;
        #pragma unroll
        for (int g = 0; g < 8; ++g) M = fmaxf(M, partial[(r * 8 + g) * 2]);
        float S = 0.f;
        #pragma unroll
        for (int g = 0; g < 8; ++g)
            S += partial[(r * 8 + g) * 2 + 1] * __expf(partial[(r * 8 + g) * 2] - M);
        acc += (M + __logf(S)) - target[r];
    }
    red[threadIdx.x] = acc;
    __syncthreads();
    for (int s = 128; s >= 1; s >>= 1) {
        if (threadIdx.x < s) red[threadIdx.x] += red[threadIdx.x + s];
        __syncthreads();
    }
    if (threadIdx.x == 0) out[0] = red[0] / 2048.0f;
}

// ---------------------------------------------------------------------------
extern "C" void kernel_launch(void* const* d_in, const int* in_sizes, int n_in,
                              void* d_out, int out_size, void* d_ws, size_t ws_size,
                              hipStream_t stream)
{
    const float* rep  = (const float*)d_in[0];  // representations (2048,64,512)
    const int*   lens = (const int*)d_in[1];    // input_lengths (2048,)
    const float* W    = (const float*)d_in[2];  // proj_w (256,512)
    const float* bias = (const float*)d_in[3];  // proj_b (256,)
    float* out = (float*)d_out;

    float* Z       = (float*)d_ws;              // 2048*256 floats (2 MB)
    float* target  = Z + (size_t)N_ROWS * P_DIM;            // 2048 floats
    float* partial = target + N_ROWS;                       // 2048*8*2 floats

    proj_kernel<<<dim3(128, 16), 32, 0, stream>>>(rep, lens, W, bias, Z);
    norm_kernel<<<256, 256, 0, stream>>>(Z);
    sim_lse_kernel<<<dim3(128, 8), 32, 0, stream>>>(Z, partial, target);
    finalize_kernel<<<1, 256, 0, stream>>>(partial, target, out);
}